// RTDETRPostProcessor_51926154608686
// MI455X (gfx1250) — compile-verified
//
#include <hip/hip_runtime.h>
#include <hip/hip_bf16.h>

// RT-DETR post-processing for MI455X (gfx1250, wave32).
// One 1024-thread workgroup per image (grid = 128).
//
//  1) Pass 0: float4-vectorized stream of the logits, per-wave-privatized
//     256-bin MSB histogram (sigmoid is monotonic -> select on raw bits).
//  2) Compact every element whose MSB digit >= threshold digit into a
//     64 KB LDS candidate list (expected ~3.9k of 160k; uniform fallback to
//     global re-reads if it ever overflows). Remaining 3 radix passes and
//     the final collection then run entirely out of LDS.
//  3) Bitonic-sort 1024 packed (~key<<32|index) entries == jax top_k order.
//  4) Gather the 300 winner boxes with per-lane async global->LDS B128 loads
//     (ASYNCcnt path), convert cxcywh -> xyxy and scale.
//  5) 300x300 IoU -> 300x10 suppression bit-matrix in LDS; greedy keep loop
//     on a single lane; write labels/boxes/scores/keep as flat float32.

#define NB    128
#define NQ    2000
#define NC    80
#define QCN   (NQ * NC)       // 160000 flat scores per image
#define KSEL  300
#define NT    1024
#define WAVES (NT / 32)
#define CAP   1024            // final candidate buffer (>= KSEL + tie slack)
#define CAP2  8192            // MSB-bin compaction buffer (64 KB)
#define NWRD  10              // ceil(300/32) keep-mask words

typedef unsigned long long ull;

__device__ __forceinline__ unsigned f2key(float f) {
    // Order-preserving float -> uint map (descending float == descending uint).
    unsigned u = __float_as_uint(f);
    return u ^ ((unsigned)((int)u >> 31) | 0x80000000u);
}

__device__ __forceinline__ float key2f(unsigned k) {
    unsigned u = (k & 0x80000000u) ? (k ^ 0x80000000u) : ~k;
    return __uint_as_float(u);
}

__global__ __launch_bounds__(NT) void rtdetr_post_kernel(
    const float* __restrict__ logits,    // [B, Q, C]
    const float* __restrict__ boxes,     // [B, Q, 4] cxcywh
    const int*   __restrict__ sizes,     // [B, 2]
    float*       __restrict__ out)       // labels|boxes|scores|keep (flat f32)
{
    __shared__ unsigned histW[WAVES][256];   // per-wave privatized pass-0 hist
    __shared__ unsigned histT[256];
    __shared__ unsigned sPrefix, sKRem, sCnt, sCnt2, sD0;
    __shared__ ull      cand2[CAP2];         // MSB-bin compaction
    __shared__ ull      cand[CAP];
    __shared__ __align__(16) float boxRaw[KSEL][4];  // async-load target
    __shared__ float boxXY[KSEL][5];         // stride-5 padding, bank-friendly
    __shared__ float areas[KSEL];
    __shared__ unsigned supMask[KSEL][NWRD];
    __shared__ unsigned keepW[NWRD];

    const int b   = blockIdx.x;
    const int tid = threadIdx.x;
    const int wv  = tid >> 5;
    const float*  lg  = logits + (size_t)b * QCN;
    const float4* lg4 = (const float4*)lg;

    // ---- 1) pass 0: MSB histogram, float4 stream, per-wave bins -------------
    for (int i = tid; i < WAVES * 256; i += NT) ((unsigned*)histW)[i] = 0u;
    if (tid == 0) { sKRem = KSEL; sCnt2 = 0u; }
    __syncthreads();
    for (int i = tid; i < QCN / 4; i += NT) {
        float4 v = lg4[i];
        atomicAdd(&histW[wv][f2key(v.x) >> 24], 1u);
        atomicAdd(&histW[wv][f2key(v.y) >> 24], 1u);
        atomicAdd(&histW[wv][f2key(v.z) >> 24], 1u);
        atomicAdd(&histW[wv][f2key(v.w) >> 24], 1u);
    }
    __syncthreads();
    if (tid < 256) {
        unsigned s = 0u;
        #pragma unroll
        for (int w = 0; w < WAVES; ++w) s += histW[w][tid];
        histT[tid] = s;
    }
    __syncthreads();
    if (tid == 0) {
        unsigned kr = sKRem, cum = 0u;
        int d = 255;
        for (; d >= 0; --d) { cum += histT[d]; if (cum >= kr) break; }
        if (d < 0) d = 0;
        sKRem = kr - (cum - histT[d]);
        sD0   = (unsigned)d;
    }
    __syncthreads();
    const unsigned d0 = sD0;

    // ---- 2) compact MSB-digit >= d0 candidates into LDS ---------------------
    for (int i = tid; i < QCN / 4; i += NT) {
        float4 v = lg4[i];
        const unsigned base = (unsigned)i * 4u;
        float e[4] = {v.x, v.y, v.z, v.w};
        #pragma unroll
        for (int c = 0; c < 4; ++c) {
            unsigned key = f2key(e[c]);
            if ((key >> 24) >= d0) {
                unsigned slot = atomicAdd(&sCnt2, 1u);
                if (slot < CAP2)
                    cand2[slot] = ((ull)(~key) << 32) | (base + c);
            }
        }
    }
    __syncthreads();
    const unsigned nC     = sCnt2;
    const bool     useLds = (nC <= CAP2);    // uniform across the block

    unsigned prefix = d0 << 24, pmask = 0xFF000000u;

    // ---- 3) radix passes 1..3 on the compact list (or global fallback) ------
    for (int pass = 1; pass < 4; ++pass) {
        const int shift = 24 - 8 * pass;
        if (tid < 256) histT[tid] = 0u;
        __syncthreads();
        if (useLds) {
            for (unsigned i = tid; i < nC; i += NT) {
                unsigned key = ~(unsigned)(cand2[i] >> 32);
                if ((key & pmask) == prefix)
                    atomicAdd(&histT[(key >> shift) & 0xFFu], 1u);
            }
        } else {
            for (int i = tid; i < QCN; i += NT) {
                unsigned key = f2key(lg[i]);
                if ((key & pmask) == prefix)
                    atomicAdd(&histT[(key >> shift) & 0xFFu], 1u);
            }
        }
        __syncthreads();
        if (tid == 0) {
            unsigned kr = sKRem, cum = 0u;
            int d = 255;
            for (; d >= 0; --d) { cum += histT[d]; if (cum >= kr) break; }
            if (d < 0) d = 0;
            sKRem   = kr - (cum - histT[d]);
            sPrefix = prefix | ((unsigned)d << shift);
        }
        __syncthreads();
        prefix = sPrefix;
        pmask |= (0xFFu << shift);
    }
    // prefix == exact key of the K-th element (threshold T)

    // ---- 4) collect key >= T, bitonic sort to jax top_k order ---------------
    if (tid == 0) sCnt = 0u;
    for (int i = tid; i < CAP; i += NT) cand[i] = ~0ULL;   // pad sorts last
    __syncthreads();
    if (useLds) {
        for (unsigned i = tid; i < nC; i += NT) {
            ull e = cand2[i];
            unsigned key = ~(unsigned)(e >> 32);
            if (key >= prefix) {
                unsigned slot = atomicAdd(&sCnt, 1u);
                if (slot < CAP) cand[slot] = e;
            }
        }
    } else {
        for (int i = tid; i < QCN; i += NT) {
            unsigned key = f2key(lg[i]);
            if (key >= prefix) {
                unsigned slot = atomicAdd(&sCnt, 1u);
                if (slot < CAP)
                    cand[slot] = ((ull)(~key) << 32) | (unsigned)i;
            }
        }
    }
    __syncthreads();
    // ascending sort on (~key, index): == (score desc, index asc)
    for (unsigned kk = 2; kk <= CAP; kk <<= 1) {
        for (unsigned j = kk >> 1; j > 0; j >>= 1) {
            unsigned ixj = (unsigned)tid ^ j;
            if (ixj > (unsigned)tid) {
                ull a = cand[tid], c = cand[ixj];
                bool up = ((tid & kk) == 0);
                if ((a > c) == up) { cand[tid] = c; cand[ixj] = a; }
            }
            __syncthreads();
        }
    }

    // ---- 5) decode winners + async gather of their boxes into LDS -----------
    float    sc    = 0.0f;
    unsigned label = 0u;
    if (tid < KSEL) {
        ull e = cand[tid];
        unsigned idx = (unsigned)e;
        unsigned key = ~(unsigned)(e >> 32);
        sc    = 1.0f / (1.0f + __expf(-key2f(key)));   // sigmoid on winners only
        label = idx % NC;
        unsigned qi = idx / NC;

        unsigned lds = (unsigned)(ull)(&boxRaw[tid][0]);
        ull ga = (ull)(boxes + ((size_t)b * NQ + qi) * 4);
        // CDNA5 async global->LDS copy, tracked by ASYNCcnt (per-lane gather).
        asm volatile("global_load_async_to_lds_b128 %0, %1, off"
                     :: "v"(lds), "v"(ga) : "memory");
    }
    asm volatile("s_wait_asynccnt 0" ::: "memory");
    __syncthreads();

    const float s0 = (float)sizes[b * 2 + 0];
    const float s1 = (float)sizes[b * 2 + 1];
    if (tid < KSEL) {
        float cx = boxRaw[tid][0], cy = boxRaw[tid][1];
        float w  = boxRaw[tid][2], h  = boxRaw[tid][3];
        float x1 = (cx - 0.5f * w) * s0;
        float y1 = (cy - 0.5f * h) * s1;
        float x2 = (cx + 0.5f * w) * s0;
        float y2 = (cy + 0.5f * h) * s1;
        boxXY[tid][0] = x1; boxXY[tid][1] = y1;
        boxXY[tid][2] = x2; boxXY[tid][3] = y2;
        areas[tid] = (x2 - x1) * (y2 - y1);
    }
    __syncthreads();

    // ---- 6) suppression bit-matrix + greedy NMS -----------------------------
    for (int t = tid; t < KSEL * NWRD; t += NT) {
        const int i = t / NWRD, w = t % NWRD;
        const float ax1 = boxXY[i][0], ay1 = boxXY[i][1];
        const float ax2 = boxXY[i][2], ay2 = boxXY[i][3];
        const float aa  = areas[i];
        unsigned bits = 0u;
        const int j0 = w * 32;
        const int j1 = (j0 + 32 < KSEL) ? (j0 + 32) : KSEL;
        for (int j = j0; j < j1; ++j) {
            float lx = fmaxf(ax1, boxXY[j][0]);
            float ly = fmaxf(ay1, boxXY[j][1]);
            float rx = fminf(ax2, boxXY[j][2]);
            float ry = fminf(ay2, boxXY[j][3]);
            float iw = fmaxf(rx - lx, 0.0f);
            float ih = fmaxf(ry - ly, 0.0f);
            float inter = iw * ih;
            float denom = aa + areas[j] - inter + 1e-9f;
            if (inter > 0.5f * denom)              // iou > 0.5
                bits |= 1u << (j - j0);
        }
        supMask[i][w] = bits;
    }
    __syncthreads();

    if (tid == 0) {
        unsigned kw[NWRD];
        #pragma unroll
        for (int w = 0; w < NWRD; ++w) kw[w] = 0u;
        for (int i = 0; i < KSEL; ++i) {
            const int wi = i >> 5;
            unsigned sup = 0u;
            for (int w = 0; w <= wi; ++w) {
                unsigned m = supMask[i][w] & kw[w];
                if (w == wi) m &= ((1u << (i & 31)) - 1u);   // only j < i
                sup |= m;
            }
            if (sup == 0u) kw[wi] |= 1u << (i & 31);
        }
        #pragma unroll
        for (int w = 0; w < NWRD; ++w) keepW[w] = kw[w];
    }
    __syncthreads();

    // ---- 7) outputs: labels | boxes | scores | keep (flat float32) ----------
    if (tid < KSEL) {
        const size_t base = (size_t)b * KSEL + tid;
        out[base] = (float)label;                                  // labels [B,K]
        float* ob = out + (size_t)NB * KSEL + base * 4;            // boxes  [B,K,4]
        ob[0] = boxXY[tid][0]; ob[1] = boxXY[tid][1];
        ob[2] = boxXY[tid][2]; ob[3] = boxXY[tid][3];
        out[(size_t)NB * KSEL * 5 + base] = sc;                    // scores [B,K]
        out[(size_t)NB * KSEL * 6 + base] =                        // keep   [B,K]
            ((keepW[tid >> 5] >> (tid & 31)) & 1u) ? 1.0f : 0.0f;
    }
}

extern "C" void kernel_launch(void* const* d_in, const int* in_sizes, int n_in,
                              void* d_out, int out_size, void* d_ws, size_t ws_size,
                              hipStream_t stream) {
    (void)in_sizes; (void)n_in; (void)d_ws; (void)ws_size; (void)out_size;
    const float* logits = (const float*)d_in[0];
    const float* boxes  = (const float*)d_in[1];
    const int*   sizes  = (const int*)d_in[2];
    float* out = (float*)d_out;
    rtdetr_post_kernel<<<NB, NT, 0, stream>>>(logits, boxes, sizes, out);
}